// GraphConvolutionII_62878321213495
// MI455X (gfx1250) — compile-verified
//
#include <hip/hip_runtime.h>

typedef __attribute__((ext_vector_type(2))) float v2f;
typedef __attribute__((ext_vector_type(8))) float v8f;
typedef __attribute__((ext_vector_type(4))) unsigned int u32x4;
typedef __attribute__((ext_vector_type(4))) int i32x4;
typedef __attribute__((ext_vector_type(8))) int i32x8;

#define DIM 256
#define ROWS_PER_BLOCK 128
#define THREADS 512
#define S_STRIDE 260   // support LDS row stride (floats): conflict-free b64 A-frags
#define KCHUNK 64
#define W_STRIDE 264   // row-major W chunk stride (floats) == TDM pad 8 dwords / 256

#ifndef __has_builtin
#define __has_builtin(x) 0
#endif
#if __has_builtin(__builtin_amdgcn_tensor_load_to_lds)
#define HAVE_TDM 1
#else
#define HAVE_TDM 0
#endif

// ---------------------------------------------------------------------------
// Kernel 1: hi[dst] += val * x[src] (scatter-add SpMM). One thread per
// (edge, float4); 32 lanes cover 512 contiguous bytes of one source row
// (coalesced, L2-resident), scatter via hardware global_atomic_add_f32.
// ---------------------------------------------------------------------------
__global__ void spmm_scatter_kernel(const float* __restrict__ x,
                                    const float* __restrict__ edge_val,
                                    const int* __restrict__ edge_src,
                                    const int* __restrict__ edge_dst,
                                    float* __restrict__ hi,
                                    long long total4) {
  long long gid = (long long)blockIdx.x * blockDim.x + threadIdx.x;
  if (gid >= total4) return;
  int e = (int)(gid >> 6);
  int q = (int)(gid & 63);
  int s = edge_src[e];
  int t = edge_dst[e];
  float v = edge_val[e];
  const float4* x4 = (const float4*)x;
  float4 xv = x4[(long long)s * (DIM / 4) + q];
  float* dst = hi + (long long)t * DIM + q * 4;
  unsafeAtomicAdd(dst + 0, v * xv.x);
  unsafeAtomicAdd(dst + 1, v * xv.y);
  unsafeAtomicAdd(dst + 2, v * xv.z);
  unsafeAtomicAdd(dst + 3, v * xv.w);
}

// ---------------------------------------------------------------------------
// Kernel 2: support = (1-a)*hi + a*h0 staged in LDS, W chunks staged via the
// Tensor Data Mover (with LDS pad -> stride 264), GEMM via
// v_wmma_f32_16x16x4_f32, fused epilogue theta*acc + (1-theta)*support + x.
// `io` holds hi on entry and the output on exit (block-private rows).
// ---------------------------------------------------------------------------
__global__ void gcnii_wmma_kernel(const float* __restrict__ x,
                                  const float* __restrict__ h0,
                                  const float* __restrict__ W,
                                  const float* __restrict__ lamda_p,
                                  const float* __restrict__ alpha_p,
                                  const int* __restrict__ l_p,
                                  float* __restrict__ io,
                                  int N) {
  extern __shared__ float smem[];
  float* sS = smem;                                 // ROWS_PER_BLOCK x S_STRIDE
  float* sW = smem + ROWS_PER_BLOCK * S_STRIDE;     // KCHUNK x W_STRIDE (row-major)

  const int tid = threadIdx.x;
  const int wid = tid >> 5;              // 16 waves
  const int lane = tid & 31;
  const int rowbase = blockIdx.x * ROWS_PER_BLOCK;

  const float alpha = *alpha_p;
  const float lam = *lamda_p;
  const float lf = (float)(*l_p);
  const float theta = __logf(lam / lf + 1.0f);
  const float om_alpha = 1.0f - alpha;
  const float om_theta = 1.0f - theta;

  // ---- Stage support tile (zeros past N) ----
  const float4* hi4 = (const float4*)io;
  const float4* h04 = (const float4*)h0;
#pragma unroll 1
  for (int i = 0; i < (ROWS_PER_BLOCK * DIM / 4) / THREADS; ++i) {  // 16 iters
    int eid = i * THREADS + tid;
    int r = eid >> 6;
    int c4 = eid & 63;
    int g = rowbase + r;
    float4 sv = make_float4(0.f, 0.f, 0.f, 0.f);
    if (g < N) {
      float4 a = hi4[(long long)g * (DIM / 4) + c4];
      float4 b = h04[(long long)g * (DIM / 4) + c4];
      sv.x = om_alpha * a.x + alpha * b.x;
      sv.y = om_alpha * a.y + alpha * b.y;
      sv.z = om_alpha * a.z + alpha * b.z;
      sv.w = om_alpha * a.w + alpha * b.w;
    }
    *(float4*)&sS[r * S_STRIDE + c4 * 4] = sv;
  }

  // ---- GEMM: wave pair (wid>>1) owns 16 rows; (wid&1) selects 8 col-tiles ---
  v8f acc[8] = {};
  const int rowtile = wid >> 1;
  const int ctbase = (wid & 1) * 8;
  const int mrow = rowtile * 16 + (lane & 15);  // A-frag row (M = lane%16)
  const int khalf = (lane >> 4) * 2;            // lanes 16-31 hold K+2,K+3
  const unsigned sW_lds_bytes = (unsigned)(uintptr_t)(void*)sW;  // LDS byte offset

#pragma unroll 1
  for (int kc = 0; kc < DIM / KCHUNK; ++kc) {
    __syncthreads();  // support staged / previous W chunk consumed
#if HAVE_TDM
    if (wid == 0) {
      // D# group0: count=1, lds_addr, 57-bit global addr, type=2 ("image")
      unsigned long long ga =
          (unsigned long long)(uintptr_t)(W + (size_t)kc * KCHUNK * DIM);
      u32x4 g0 = {1u, sW_lds_bytes, (unsigned)(ga & 0xffffffffu),
                  (unsigned)((ga >> 32) & 0x01ffffffu) | (2u << 30)};
      // D# group1: data_size=4B(2), pad_enable, pad_interval=7 (256 dw),
      // pad_amount=7 (8 dw) -> LDS stride 264 floats; tensor 256x256,
      // tile 256x64, dim0_stride=256.
      i32x8 g1 = {(2 << 16) | (1 << 20) | (7 << 22) | (7 << 25),
                  (int)(256u << 16),   // tensor_dim0[15:0]=256 in [31:16]
                  (int)(256u << 16),   // tensor_dim1[15:0]=256 in [31:16]
                  (int)(256u << 16),   // tile_dim0=256 in [31:16]
                  KCHUNK,              // tile_dim1=64
                  DIM,                 // tensor_dim0_stride=256
                  0, 0};
      i32x4 z4 = {0, 0, 0, 0};
#if __clang_major__ >= 23
      i32x8 z8 = {0, 0, 0, 0, 0, 0, 0, 0};
      __builtin_amdgcn_tensor_load_to_lds(g0, g1, z4, z4, z8, 0);
#else
      __builtin_amdgcn_tensor_load_to_lds(g0, g1, z4, z4, 0);
#endif
      __builtin_amdgcn_s_wait_tensorcnt(0);
    }
#else
    // Fallback: cooperative row-major staging of the W chunk
#pragma unroll 1
    for (int i = 0; i < KCHUNK * DIM / THREADS; ++i) {  // 32 iters
      int eid = i * THREADS + tid;
      int kk = eid >> 8;
      int n = eid & 255;
      sW[kk * W_STRIDE + n] = W[(long long)(kc * KCHUNK + kk) * DIM + n];
    }
#endif
    __syncthreads();

#pragma unroll 1
    for (int ks = 0; ks < KCHUNK; ks += 4) {
      // Batch all fragment loads into distinct registers, then 8 WMMAs.
      v2f a = *(const v2f*)&sS[mrow * S_STRIDE + kc * KCHUNK + ks + khalf];
      v2f b[8];
#pragma unroll
      for (int ci = 0; ci < 8; ++ci) {
        int col = (ctbase + ci) * 16 + (lane & 15);
        b[ci].x = sW[(ks + khalf) * W_STRIDE + col];
        b[ci].y = sW[(ks + khalf + 1) * W_STRIDE + col];
      }
#pragma unroll
      for (int ci = 0; ci < 8; ++ci) {
        acc[ci] = __builtin_amdgcn_wmma_f32_16x16x4_f32(
            false, a, false, b[ci], (short)0, acc[ci], false, false);
      }
    }
  }

  // ---- Epilogue: out = theta*acc + (1-theta)*support + x ----
  // C/D layout: VGPR j, lanes 0-15 -> M=j, lanes 16-31 -> M=j+8, N=lane%16.
#pragma unroll
  for (int ci = 0; ci < 8; ++ci) {
#pragma unroll
    for (int j = 0; j < 8; ++j) {
      int rl = rowtile * 16 + j + ((lane >> 4) << 3);
      int col = (ctbase + ci) * 16 + (lane & 15);
      int g = rowbase + rl;
      if (g < N) {
        float sup = sS[rl * S_STRIDE + col];
        float xv = x[(long long)g * DIM + col];
        io[(long long)g * DIM + col] = theta * acc[ci][j] + om_theta * sup + xv;
      }
    }
  }
}

extern "C" void kernel_launch(void* const* d_in, const int* in_sizes, int n_in,
                              void* d_out, int out_size, void* d_ws, size_t ws_size,
                              hipStream_t stream) {
  const float* x        = (const float*)d_in[0];
  const float* h0       = (const float*)d_in[1];
  const float* edge_val = (const float*)d_in[2];
  const float* W        = (const float*)d_in[3];
  const float* lamda    = (const float*)d_in[4];
  const float* alpha    = (const float*)d_in[5];
  const int*   edge_src = (const int*)d_in[6];
  const int*   edge_dst = (const int*)d_in[7];
  const int*   l        = (const int*)d_in[8];
  float* out = (float*)d_out;

  int N = in_sizes[0] / DIM;
  int E = in_sizes[2];

  // d_out doubles as the hi accumulator; zero it first.
  hipMemsetAsync(out, 0, (size_t)N * DIM * sizeof(float), stream);

  long long total4 = (long long)E * (DIM / 4);
  int g1 = (int)((total4 + 255) / 256);
  spmm_scatter_kernel<<<g1, 256, 0, stream>>>(x, edge_val, edge_src, edge_dst,
                                              out, total4);

  int g2 = (N + ROWS_PER_BLOCK - 1) / ROWS_PER_BLOCK;
  size_t smem_bytes =
      (size_t)(ROWS_PER_BLOCK * S_STRIDE + KCHUNK * W_STRIDE) * sizeof(float);
  gcnii_wmma_kernel<<<g2, THREADS, smem_bytes, stream>>>(x, h0, W, lamda, alpha,
                                                         l, out, N);
}